// res_GAT_meta_block_5738076307728
// MI455X (gfx1250) — compile-verified
//
#include <hip/hip_runtime.h>
#include <math.h>

typedef float v2f __attribute__((ext_vector_type(2)));
typedef float v8f __attribute__((ext_vector_type(8)));

#define CC   128
#define NN   8192
#define BB   8
#define KK   16
#define LSTR 24   // LDS row stride: 2*LSTR % 32 == 16 -> half-waves hit disjoint bank halves

__device__ __forceinline__ float lrelu(float x) { return x >= 0.0f ? x : 0.2f * x; }

// ---------------- Kernel 1: hypernetwork -> W1, W2 (row-major [o][i]) ----------------
__global__ void hyper_kernel(const int* __restrict__ up_scale,
                             const float* __restrict__ w11a, const float* __restrict__ b11a,
                             const float* __restrict__ w12a, const float* __restrict__ b12a,
                             const float* __restrict__ wska, const float* __restrict__ bska,
                             const float* __restrict__ w11b, const float* __restrict__ b11b,
                             const float* __restrict__ w12b, const float* __restrict__ b12b,
                             const float* __restrict__ wskb, const float* __restrict__ bskb,
                             float* __restrict__ W)  // 2*128*128 floats
{
    int e = blockIdx.x * blockDim.x + threadIdx.x;   // 0..32767
    // scale value, computed in double exactly like the numpy reference
    double us = (double)up_scale[0];
    double u2 = us * us;
    int idx = (int)((u2 - 1.1) / 0.1);
    double up   = 1.1 + 0.1 * (double)idx;
    double last = 1.1 + 0.1 * 79.0;
    float s = (float)(sqrt(up) / sqrt(last));

    int tag = e >> 14, eid = e & 16383;
    const float *w11, *b11, *w12, *b12, *wsk, *bsk;
    if (tag == 0) { w11 = w11a; b11 = b11a; w12 = w12a; b12 = b12a; wsk = wska; bsk = bska; }
    else          { w11 = w11b; b11 = b11b; w12 = w12b; b12 = b12b; wsk = wskb; bsk = bskb; }

    float acc = b12[eid] + wsk[eid] * s + bsk[eid];
    const float* wrow = w12 + (size_t)eid * 32;
#pragma unroll
    for (int j = 0; j < 32; ++j) {
        float h = lrelu(w11[j] * s + b11[j]);
        acc += wrow[j] * h;
    }
    W[e] = acc;
}

// ---------------- Kernel 2: reg_loss = 0.5 * sum(W^2), deterministic single block ----
__global__ void reg_kernel(const float* __restrict__ W, float* __restrict__ out_reg)
{
    __shared__ float red[256];
    float sum = 0.0f;
    for (int i = threadIdx.x; i < 2 * CC * CC; i += 256) { float v = W[i]; sum += v * v; }
    red[threadIdx.x] = sum;
    __syncthreads();
    for (int s = 128; s > 0; s >>= 1) {
        if ((int)threadIdx.x < s) red[threadIdx.x] += red[threadIdx.x + s];
        __syncthreads();
    }
    if (threadIdx.x == 0) out_reg[0] = 0.5f * red[0];
}

// ---------------- Kernel 3: yt[b][n][c] = lrelu(points[b][c][n]) (LDS tiled) ---------
__global__ __launch_bounds__(256) void act_transpose_kernel(const float* __restrict__ points,
                                                            float* __restrict__ yt)
{
    __shared__ float tile[32][33];
    int b = blockIdx.z, c0 = blockIdx.y * 32, n0 = blockIdx.x * 32;
    int tn = threadIdx.x & 31, tr = threadIdx.x >> 5;   // tr: 0..7
    const float* src = points + (size_t)b * CC * NN;
#pragma unroll
    for (int p = 0; p < 4; ++p) {
        int c = tr + p * 8;
        tile[c][tn] = lrelu(src[(size_t)(c0 + c) * NN + n0 + tn]);
    }
    __syncthreads();
    float* dst = yt + (size_t)b * NN * CC;
#pragma unroll
    for (int p = 0; p < 4; ++p) {
        int n = tr + p * 8;
        dst[(size_t)(n0 + n) * CC + c0 + tn] = tile[tn][n];
    }
}

// ---------------- Kernel 4: main fused gather + dual GEMM + epilogue -----------------
__global__ __launch_bounds__(256) void meta_gemm_kernel(const float* __restrict__ yt,
                                                        const float* __restrict__ W1,
                                                        const float* __restrict__ W2,
                                                        const int* __restrict__ indices,
                                                        float* __restrict__ out)
{
    __shared__ float xs[CC * LSTR];   // x tile, [c][n] stride LSTR
    __shared__ float gs[CC * LSTR];   // K-summed gather tile

    int blk = blockIdx.x;
    int b   = blk >> 9;               // NN/16 = 512 tiles per batch
    int n0  = (blk & 511) << 4;
    int tid = threadIdx.x;
    int nl  = tid >> 4;               // 0..15 : column within tile
    int l   = tid & 15;               // 0..15 : 8-float channel chunk

    const float* ybase = yt + (size_t)b * NN * CC;

    // stage xs: thread covers (n = nl, c = 8l..8l+7); 16 lanes read one 512B row
    {
        const float4* r = (const float4*)(ybase + (size_t)(n0 + nl) * CC + l * 8);
        float4 v0 = r[0], v1 = r[1];
        float vv[8] = { v0.x, v0.y, v0.z, v0.w, v1.x, v1.y, v1.z, v1.w };
#pragma unroll
        for (int j = 0; j < 8; ++j) xs[(l * 8 + j) * LSTR + nl] = vv[j];
    }

    // gather + K-sum: 16 coalesced 512B row reads per column, all L2-resident
    {
        float a[8] = {0, 0, 0, 0, 0, 0, 0, 0};
        const int* irow = indices + ((size_t)b * NN + n0 + nl) * KK;
#pragma unroll 4
        for (int k = 0; k < KK; ++k) {
            int id = irow[k];
            const float4* r = (const float4*)(ybase + (size_t)id * CC + l * 8);
            float4 v0 = r[0], v1 = r[1];
            a[0] += v0.x; a[1] += v0.y; a[2] += v0.z; a[3] += v0.w;
            a[4] += v1.x; a[5] += v1.y; a[6] += v1.z; a[7] += v1.w;
        }
#pragma unroll
        for (int j = 0; j < 8; ++j) gs[(l * 8 + j) * LSTR + nl] = a[j];
    }
    __syncthreads();

    // WMMA: 8 waves, one 16-row output tile each; 64 chained f32 16x16x4 WMMAs
    int lane = tid & 31;
    int wv   = tid >> 5;
    int o0   = wv * 16;
    int m    = lane & 15;
    int kb   = (lane >> 4) * 2;       // A/B K-split per half-wave
    int bn   = lane & 15;

    const float* w1row = W1 + (size_t)(o0 + m) * CC + kb;
    const float* w2row = W2 + (size_t)(o0 + m) * CC + kb;

    v8f acc = {};
#pragma unroll 8
    for (int i0 = 0; i0 < CC; i0 += 4) {
        v2f a1 = { w1row[i0], w1row[i0 + 1] };
        v2f bx = { xs[(i0 + kb) * LSTR + bn], xs[(i0 + kb + 1) * LSTR + bn] };
        acc = __builtin_amdgcn_wmma_f32_16x16x4_f32(false, a1, false, bx, (short)0, acc, false, false);
        v2f a2 = { w2row[i0], w2row[i0 + 1] };
        v2f bg = { gs[(i0 + kb) * LSTR + bn], gs[(i0 + kb + 1) * LSTR + bn] };
        acc = __builtin_amdgcn_wmma_f32_16x16x4_f32(false, a2, false, bg, (short)0, acc, false, false);
    }

    // epilogue: /(K+1) + x, coalesced 64B stores per half-wave row
    int hi = lane >> 4;
    const float inv = 1.0f / 17.0f;
    float* orow = out + (size_t)b * CC * NN + n0 + bn;
#pragma unroll
    for (int r = 0; r < 8; ++r) {
        int o = o0 + r + 8 * hi;
        orow[(size_t)o * NN] = acc[r] * inv + xs[o * LSTR + bn];
    }
}

// ---------------- host launcher ------------------------------------------------------
extern "C" void kernel_launch(void* const* d_in, const int* in_sizes, int n_in,
                              void* d_out, int out_size, void* d_ws, size_t ws_size,
                              hipStream_t stream)
{
    (void)in_sizes; (void)n_in; (void)out_size; (void)ws_size;

    const float* points  = (const float*)d_in[0];
    const int*   indices = (const int*)d_in[1];
    const int*   upsc    = (const int*)d_in[2];
    const float* w11a = (const float*)d_in[3],  *b11a = (const float*)d_in[4];
    const float* w12a = (const float*)d_in[5],  *b12a = (const float*)d_in[6];
    const float* wska = (const float*)d_in[7],  *bska = (const float*)d_in[8];
    const float* w11b = (const float*)d_in[9],  *b11b = (const float*)d_in[10];
    const float* w12b = (const float*)d_in[11], *b12b = (const float*)d_in[12];
    const float* wskb = (const float*)d_in[13], *bskb = (const float*)d_in[14];

    float* out = (float*)d_out;                       // [8*128*8192] features + [1] reg
    float* W   = (float*)d_ws;                        // 32768 floats: W1 then W2
    float* yt  = W + 2 * CC * CC;                     // 8388608 floats: lrelu(points)^T

    hyper_kernel<<<128, 256, 0, stream>>>(upsc,
        w11a, b11a, w12a, b12a, wska, bska,
        w11b, b11b, w12b, b12b, wskb, bskb, W);

    reg_kernel<<<1, 256, 0, stream>>>(W, out + (size_t)BB * CC * NN);

    dim3 g2(NN / 32, CC / 32, BB);
    act_transpose_kernel<<<g2, 256, 0, stream>>>(points, yt);

    meta_gemm_kernel<<<BB * (NN / 16), 256, 0, stream>>>(yt, W, W + CC * CC, indices, out);
}